// EdgeNNConv_9672266350626
// MI455X (gfx1250) — compile-verified
//
#include <hip/hip_runtime.h>
#include <math.h>

// Problem constants (from the reference)
#define ATTR   16
#define H1DIM  256
#define H2DIM  1024
#define INC    32
#define OUTC   32
#define TILE_E 16

typedef __attribute__((ext_vector_type(2))) float v2f;
typedef __attribute__((ext_vector_type(8))) float v8f;

// Branchless ELU using the hardware transcendental (v_exp_f32).
__device__ __forceinline__ float elu1(float v) {
    const float ev = __expf(v) - 1.0f;
    return (v > 0.0f) ? v : ev;
}

// ---------------------------------------------------------------------------
// Kernel 1: out[n][o] = bias[o] + sum_i x[n][i] * root[i][o]
// ---------------------------------------------------------------------------
__global__ __launch_bounds__(256) void root_bias_kernel(
    const float* __restrict__ x, const float* __restrict__ root,
    const float* __restrict__ bias, float* __restrict__ out, int nNodes)
{
    __shared__ float rs[INC * OUTC];
    __shared__ float bs[OUTC];
    const int t = threadIdx.x;
    for (int idx = t; idx < INC * OUTC; idx += 256) rs[idx] = root[idx];
    if (t < OUTC) bs[t] = bias[t];
    __syncthreads();

    const int gid = blockIdx.x * 256 + t;
    const int n = gid >> 5;
    const int o = gid & 31;
    if (n < nNodes) {
        float acc = bs[o];
        const float* xr = x + n * INC;
        #pragma unroll
        for (int i = 0; i < INC; ++i) acc = fmaf(xr[i], rs[i * OUTC + o], acc);
        out[gid] = acc;
    }
}

// ---------------------------------------------------------------------------
// Kernel 2: fused edge MLP (WMMA f32) + batched matvec + scatter-add
// One workgroup = 8 wave32 handles TILE_E=16 edges.
// ---------------------------------------------------------------------------
__global__ __launch_bounds__(256) void edge_nnconv_kernel(
    const float* __restrict__ x, const int* __restrict__ edge_index,
    const float* __restrict__ edge_attr,
    const float* __restrict__ W1, const float* __restrict__ b1,
    const float* __restrict__ W2, const float* __restrict__ b2,
    const float* __restrict__ W3, const float* __restrict__ b3,
    float* __restrict__ out, int nEdges)
{
    __shared__ float ea [TILE_E * ATTR];    //  1 KB  edge_attr tile (row major)
    __shared__ float h1s[TILE_E * H1DIM];   // 16 KB  layer-1 activations
    __shared__ float h2s[TILE_E * H2DIM];   // 64 KB  layer-2 activations
    __shared__ float xt [TILE_E * INC];     //  2 KB  gathered x[src] tile
    __shared__ float msg[TILE_E * OUTC];    //  2 KB  per-edge message accum

    const int t    = threadIdx.x;
    const int wave = t >> 5;
    const int lane = t & 31;
    const int m    = lane & 15;             // A row / B,D column within tile
    const int koff = (lane >> 4) << 1;      // 0 (lanes 0-15) or 2 (lanes 16-31)
    const int rhi  = (lane >> 4) << 3;      // +8 row offset for upper half-wave
    const int e0   = blockIdx.x * TILE_E;

    // ---- stage inputs -----------------------------------------------------
    {
        const int el = t >> 4, a = t & 15;
        const int e  = e0 + el;
        ea[t] = (e < nEdges) ? edge_attr[e * ATTR + a] : 0.0f;
    }
    #pragma unroll
    for (int r = 0; r < 2; ++r) {
        const int idx = t + 256 * r;               // 512 = 16 edges * 32 chans
        const int el = idx >> 5, i = idx & 31;
        const int e  = e0 + el;
        const int s  = (e < nEdges) ? edge_index[e] : 0;   // src row
        xt[idx]  = x[s * INC + i];
        msg[idx] = 0.0f;
    }
    __syncthreads();

    // ---- layer 1: h1 = elu(ea @ W1 + b1)   M=16 K=16 N=256 ---------------
    // 16 N-tiles; 2 per wave, both accumulated simultaneously (shared A).
    {
        v8f c[2] = {};
        const int nb = wave * 32;                  // column base for this wave
        #pragma unroll
        for (int k = 0; k < ATTR; k += 4) {
            v2f a = *(const v2f*)&ea[m * ATTR + k + koff];
            #pragma unroll
            for (int j = 0; j < 2; ++j) {
                const int n0 = nb + j * 16;
                v2f b;
                b.x = W1[(k + koff)     * H1DIM + n0 + m];
                b.y = W1[(k + koff + 1) * H1DIM + n0 + m];
                c[j] = __builtin_amdgcn_wmma_f32_16x16x4_f32(
                    false, a, false, b, (short)0, c[j], false, false);
            }
        }
        #pragma unroll
        for (int j = 0; j < 2; ++j) {
            const int n0 = nb + j * 16;
            const float bv = b1[n0 + m];
            #pragma unroll
            for (int v = 0; v < 8; ++v)
                h1s[(v + rhi) * H1DIM + n0 + m] = elu1(c[j][v] + bv);
        }
    }
    __syncthreads();

    // ---- layer 2: h2 = elu(h1 @ W2 + b2)   M=16 K=256 N=1024 -------------
    // 64 N-tiles; 8 per wave, K-outer with 8 accumulators: one ds_load_b64 of
    // A feeds 8 back-to-back WMMAs.
    {
        v8f c[8] = {};
        const int nb = wave * 128;
        #pragma unroll 2
        for (int k = 0; k < H1DIM; k += 4) {
            v2f a = *(const v2f*)&h1s[m * H1DIM + k + koff];
            #pragma unroll
            for (int j = 0; j < 8; ++j) {
                const int n0 = nb + j * 16;
                v2f b;
                b.x = W2[(k + koff)     * H2DIM + n0 + m];
                b.y = W2[(k + koff + 1) * H2DIM + n0 + m];
                c[j] = __builtin_amdgcn_wmma_f32_16x16x4_f32(
                    false, a, false, b, (short)0, c[j], false, false);
            }
        }
        #pragma unroll
        for (int j = 0; j < 8; ++j) {
            const int n0 = nb + j * 16;
            const float bv = b2[n0 + m];
            #pragma unroll
            for (int v = 0; v < 8; ++v)
                h2s[(v + rhi) * H2DIM + n0 + m] = elu1(c[j][v] + bv);
        }
    }
    __syncthreads();

    // ---- layer 3 fused with per-edge matvec ------------------------------
    // w[e] = elu(h2 @ W3 + b3) reshaped [32 in][32 out]; column block
    // n = i*2 + hh (hh = o-half). Wave handles o-half (wave&1) and the 8
    // input channels i = (wave>>1)*8 .. +7 as 8 simultaneous accumulators
    // (shared A fragment), then folds each tile into the msg accumulator:
    // msg[row][hh*16+m] += x_src[row][i] * w_tile.
    {
        const int hh = wave & 1;
        const int ib = wave >> 1;
        v8f c[8] = {};
        #pragma unroll 2
        for (int k = 0; k < H2DIM; k += 4) {
            v2f a = *(const v2f*)&h2s[m * H2DIM + k + koff];
            #pragma unroll
            for (int ti = 0; ti < 8; ++ti) {
                const int n0 = ((ib * 8 + ti) * 2 + hh) * 16;
                v2f b;
                b.x = W3[(k + koff)     * (INC * OUTC) + n0 + m];
                b.y = W3[(k + koff + 1) * (INC * OUTC) + n0 + m];
                c[ti] = __builtin_amdgcn_wmma_f32_16x16x4_f32(
                    false, a, false, b, (short)0, c[ti], false, false);
            }
        }

        float msgacc[8];
        #pragma unroll
        for (int v = 0; v < 8; ++v) msgacc[v] = 0.0f;

        #pragma unroll
        for (int ti = 0; ti < 8; ++ti) {
            const int i  = ib * 8 + ti;
            const int n0 = (i * 2 + hh) * 16;
            const float bv = b3[n0 + m];
            #pragma unroll
            for (int v = 0; v < 8; ++v) {
                const float wv = elu1(c[ti][v] + bv);
                msgacc[v] = fmaf(xt[(v + rhi) * INC + i], wv, msgacc[v]);
            }
        }

        // reduce the 4 waves sharing each o-half into LDS msg (ds_add_f32)
        #pragma unroll
        for (int v = 0; v < 8; ++v) {
            __hip_atomic_fetch_add(&msg[(v + rhi) * OUTC + hh * 16 + m],
                                   msgacc[v], __ATOMIC_RELAXED,
                                   __HIP_MEMORY_SCOPE_WORKGROUP);
        }
    }
    __syncthreads();

    // ---- scatter-add to out[dst] -----------------------------------------
    #pragma unroll
    for (int r = 0; r < 2; ++r) {
        const int idx = t + 256 * r;
        const int el = idx >> 5, o = idx & 31;
        const int e  = e0 + el;
        if (e < nEdges) {
            const int d = edge_index[nEdges + e];   // dst row
            __hip_atomic_fetch_add(&out[d * OUTC + o], msg[idx],
                                   __ATOMIC_RELAXED, __HIP_MEMORY_SCOPE_AGENT);
        }
    }
}

// ---------------------------------------------------------------------------
extern "C" void kernel_launch(void* const* d_in, const int* in_sizes, int n_in,
                              void* d_out, int out_size, void* d_ws, size_t ws_size,
                              hipStream_t stream)
{
    const float* x         = (const float*)d_in[0];
    const int*   edge_idx  = (const int*)  d_in[1];
    const float* edge_attr = (const float*)d_in[2];
    const float* W1        = (const float*)d_in[3];
    const float* b1        = (const float*)d_in[4];
    const float* W2        = (const float*)d_in[5];
    const float* b2        = (const float*)d_in[6];
    const float* W3        = (const float*)d_in[7];
    const float* b3        = (const float*)d_in[8];
    const float* root      = (const float*)d_in[9];
    const float* bias      = (const float*)d_in[10];
    float* out = (float*)d_out;

    const int nNodes = in_sizes[0] / INC;
    const int nEdges = in_sizes[1] / 2;

    // out = x @ root + bias   (also initializes the poisoned output buffer)
    const int rb_grid = (nNodes * OUTC + 255) / 256;
    root_bias_kernel<<<rb_grid, 256, 0, stream>>>(x, root, bias, out, nNodes);

    // fused edge MLP + matvec + scatter (accumulates into out)
    const int e_grid = (nEdges + TILE_E - 1) / TILE_E;
    edge_nnconv_kernel<<<e_grid, 256, 0, stream>>>(
        x, edge_idx, edge_attr, W1, b1, W2, b2, W3, b3, out, nEdges);
}